// soft_to_hard_quantize_56959856280055
// MI455X (gfx1250) — compile-verified
//
#include <hip/hip_runtime.h>
#include <hip/hip_bf16.h>

typedef float v2f __attribute__((ext_vector_type(2)));
typedef float v4f __attribute__((ext_vector_type(4)));
typedef float v8f __attribute__((ext_vector_type(8)));

#define WAVES_PER_BLOCK 8
#define YSTRIDE 68      // y tile rows (pad 64->68)
#define TSTRIDE 68      // ref^T rows
#define SSTRIDE 20      // scaled-ref rows: 16 dims + r2 col + 3 zeros (20*L mod 64 distinct)
#define HARDNESS 5.0f
#define LOG2E    1.4426950408889634f

__global__ __launch_bounds__(256) void stq_kernel(const float* __restrict__ x,
                                                  const float* __restrict__ ref,
                                                  float* __restrict__ out,
                                                  int ntiles)
{
    __shared__ float lds_y[WAVES_PER_BLOCK * 16 * YSTRIDE];
    __shared__ float lds_rt[16 * TSTRIDE];     // ref^T : [dim][codeword]  (GEMM2 B)
    __shared__ float lds_sr[64 * SSTRIDE];     // [m][ 10*log2e*ref | -5*log2e*r2 | 0,0,0 ]

    const int tid  = threadIdx.x;
    const int lane = tid & 31;
    const int wave = tid >> 5;
    const int tile = blockIdx.x * WAVES_PER_BLOCK + wave;
    const bool active = (tile < ntiles);

    const int hi   = lane >> 4;
    const int ln   = lane & 15;
    const int koff = hi * 2;

    // ---- per-block staging: ref^T (for GEMM2 B) ----
    #pragma unroll
    for (int i = 0; i < 4; ++i) {
        const int idx = tid + 256 * i;                       // = m*16 + d
        lds_rt[(idx & 15) * TSTRIDE + (idx >> 4)] = ref[idx];
    }
    // ---- per-block staging: scaled ref + fused r2 column (GEMM1 A) ----
    if (tid < 64) {
        const float* rp = ref + tid * 16;
        v4f q[4];
        #pragma unroll
        for (int i = 0; i < 4; ++i) q[i] = *(const v4f*)(rp + 4 * i);   // b128 loads
        float r2 = 0.0f;
        #pragma unroll
        for (int i = 0; i < 4; ++i)
            r2 += q[i][0]*q[i][0] + q[i][1]*q[i][1] + q[i][2]*q[i][2] + q[i][3]*q[i][3];
        float* sp = lds_sr + tid * SSTRIDE;
        #pragma unroll
        for (int i = 0; i < 4; ++i) {
            #pragma unroll
            for (int j = 0; j < 4; ++j)
                sp[4 * i + j] = (2.0f * HARDNESS * LOG2E) * q[i][j];
        }
        sp[16] = (-HARDNESS * LOG2E) * r2;
        sp[17] = 0.0f; sp[18] = 0.0f; sp[19] = 0.0f;
    }

    __syncthreads();    // staging visible to all waves

    float* ybase = lds_y + wave * 16 * YSTRIDE;

    if (active) {
        const float* xt = x + (size_t)tile * 256;   // 16 rows x 16 dims

        // B fragments: x^T in B-layout (identical addresses to the old A loads)
        v2f X[4];
        #pragma unroll
        for (int c = 0; c < 4; ++c)
            X[c] = *(const v2f*)(xt + ln * 16 + 4 * c + koff);    // global_load_b64

        // constant-ones B chunk for the fused r2 column (K=16..19)
        v2f bc4;
        bc4.x = hi ? 0.0f : 1.0f;
        bc4.y = 0.0f;

        // ---- GEMM1 (transposed): lg2[m][row] = log2e*(10*cross - 5*r2), direct ----
        v8f lg2[4];
        #pragma unroll
        for (int mb = 0; mb < 4; ++mb) {
            const float* ap = lds_sr + (mb * 16 + ln) * SSTRIDE + koff;
            v8f a = {};
            #pragma unroll
            for (int c = 0; c < 4; ++c) {
                v2f Af = *(const v2f*)(ap + 4 * c);               // ds_load_b64
                a = __builtin_amdgcn_wmma_f32_16x16x4_f32(false, Af, false, X[c],
                                                          (short)0, a, false, false);
            }
            v2f Ar = *(const v2f*)(ap + 16);                      // r2 column chunk
            a = __builtin_amdgcn_wmma_f32_16x16x4_f32(false, Ar, false, bc4,
                                                      (short)0, a, false, false);
            lg2[mb] = a;
        }

        // ---- softmax over all 64 codewords: register-local + ONE xor-16 shuffle ----
        float mx = lg2[0][0];
        #pragma unroll
        for (int mb = 0; mb < 4; ++mb)
            #pragma unroll
            for (int r = 0; r < 8; ++r) mx = fmaxf(mx, lg2[mb][r]);
        mx = fmaxf(mx, __shfl_xor(mx, 16, 32));

        float sum = 0.0f;
        #pragma unroll
        for (int mb = 0; mb < 4; ++mb)
            #pragma unroll
            for (int r = 0; r < 8; ++r) {
                float e = __builtin_amdgcn_exp2f(lg2[mb][r] - mx);   // v_exp_f32
                lg2[mb][r] = e;
                sum += e;
            }
        sum += __shfl_xor(sum, 16, 32);
        const float inv = 1.0f / sum;

        // ---- write normalized y to LDS: [row=ln][m = mb*16 + r + 8*hi] ----
        float* yw = ybase + ln * YSTRIDE + 8 * hi;
        #pragma unroll
        for (int mb = 0; mb < 4; ++mb)
            #pragma unroll
            for (int r = 0; r < 8; ++r)
                yw[mb * 16 + r] = lg2[mb][r] * inv;               // ds_store_b32
    }

    __syncthreads();    // y transpose visibility (cross-lane)

    if (active) {
        // ---- GEMM2: z = y (16x64) x ref (64x16); both operands from LDS ----
        v8f z = {};
        #pragma unroll
        for (int c2 = 0; c2 < 16; ++c2) {
            v2f A2 = *(const v2f*)(ybase  + ln * YSTRIDE + 4 * c2 + koff);  // ds_load_b64
            v2f B2 = *(const v2f*)(lds_rt + ln * TSTRIDE + 4 * c2 + koff);  // ds_load_b64
            z = __builtin_amdgcn_wmma_f32_16x16x4_f32(false, A2, false, B2,
                                                      (short)0, z, false, false);
        }

        // ---- store (coalesced 64B row segments per half) ----
        float* ot = out + (size_t)tile * 256;
        #pragma unroll
        for (int r = 0; r < 8; ++r)
            ot[(r + 8 * hi) * 16 + ln] = z[r];
    }
}

extern "C" void kernel_launch(void* const* d_in, const int* in_sizes, int n_in,
                              void* d_out, int out_size, void* d_ws, size_t ws_size,
                              hipStream_t stream) {
    const float* x   = (const float*)d_in[0];   // (64,8,32,32,16) f32
    const float* ref = (const float*)d_in[1];   // (64,16) f32
    float* out = (float*)d_out;

    const int ntiles = in_sizes[0] / 256;       // 16 vectors x 16 dims per tile
    const int blocks = (ntiles + WAVES_PER_BLOCK - 1) / WAVES_PER_BLOCK;

    stq_kernel<<<blocks, 32 * WAVES_PER_BLOCK, 0, stream>>>(x, ref, out, ntiles);
}